// VariationalGCNEncoder_35605278883995
// MI455X (gfx1250) — compile-verified
//
#include <hip/hip_runtime.h>
#include <hip/hip_bf16.h>

typedef float v2f __attribute__((ext_vector_type(2)));
typedef float v8f __attribute__((ext_vector_type(8)));

static constexpr int N_NODES = 50000;
static constexpr int IN_CH   = 256;
static constexpr int Z_DIM   = 64;
static constexpr int N_EDGES = 800000;
static constexpr int XW_COLS = 128;              // [W_mu | W_logstd]
static constexpr int TILES_M = N_NODES / 16;     // 3125 (exact)
static constexpr int TILES_N = XW_COLS / 16;     // 8

// ---------------------------------------------------------------------------
// Degree kernels: deg starts at 1.0 (self loop), +1 per incoming edge, then
// dis = rsqrt(deg).  deg >= 1 always, matching the reference's where-clause.
// ---------------------------------------------------------------------------
__global__ void k_init_deg(float* __restrict__ deg) {
    int i = blockIdx.x * 256 + threadIdx.x;
    if (i < N_NODES) deg[i] = 1.0f;
}

__global__ void k_count_deg(const long long* __restrict__ edge_index,
                            float* __restrict__ deg) {
    int e = blockIdx.x * 256 + threadIdx.x;
    if (e < N_EDGES) {
        int d = (int)edge_index[N_EDGES + e];   // row 1 = dst
        atomicAdd(&deg[d], 1.0f);
    }
}

__global__ void k_rsqrt_deg(float* __restrict__ deg) {
    int i = blockIdx.x * 256 + threadIdx.x;
    if (i < N_NODES) deg[i] = rsqrtf(deg[i]);
}

// ---------------------------------------------------------------------------
// GEMM: xw[50000][128] = x[50000][256] @ [W_mu | W_logstd][256][128]
// fp32 WMMA (V_WMMA_F32_16X16X4_F32) — exact fp32 numerics vs reference.
// One wave per 16x16 output tile; 8 waves (256 threads) per block.
// Grid is exact so EXEC is all ones for every WMMA.
//
// Per-lane layouts (ISA 7.12.2, 32-bit operands):
//  A 16x4 : lane L holds row M=L%16; VGPR0/1 = K = 2*(L/16) + {0,1}
//  B 4x16 : lane L holds col N=L%16; VGPR0/1 = K = 2*(L/16) + {0,1}
//  C/D    : VGPR r, lanes 0-15 -> (M=r, N=L); lanes 16-31 -> (M=8+r, N=L-16)
// ---------------------------------------------------------------------------
__global__ void k_gemm_xw_wmma(const float* __restrict__ x,
                               const float* __restrict__ Wmu,
                               const float* __restrict__ Wls,
                               float* __restrict__ xw) {
    const int lane  = threadIdx.x & 31;
    const int wave  = threadIdx.x >> 5;
    const int tile  = blockIdx.x * 8 + wave;   // 0 .. 24999
    const int tileM = tile >> 3;               // 0 .. 3124
    const int tileN = tile & 7;                // 0 .. 7

    const float* __restrict__ W = (tileN < 4) ? Wmu : Wls;
    const int colBase = (tileN & 3) * 16;

    const int m     = lane & 15;               // M (for A) / N (for B)
    const int kHalf = (lane >> 4) * 2;         // 0 or 2
    const float* __restrict__ xrow = x + (size_t)(tileM * 16 + m) * IN_CH;
    const float* __restrict__ wcol = W + (colBase + m);

    v8f acc = {};
    #pragma unroll 4
    for (int k0 = 0; k0 < IN_CH; k0 += 4) {
        v2f a, b;
        a.x = xrow[k0 + kHalf + 0];
        a.y = xrow[k0 + kHalf + 1];
        b.x = wcol[(size_t)(k0 + kHalf + 0) * Z_DIM];
        b.y = wcol[(size_t)(k0 + kHalf + 1) * Z_DIM];
        // 8 args: (neg_a, A, neg_b, B, c_mod, C, reuse_a, reuse_b)
        acc = __builtin_amdgcn_wmma_f32_16x16x4_f32(
            false, a, false, b, (short)0, acc, false, false);
    }

    const int outRowBase = tileM * 16 + (lane >> 4) * 8;
    const int outCol     = tileN * 16 + (lane & 15);
    #pragma unroll
    for (int r = 0; r < 8; ++r) {
        xw[(size_t)(outRowBase + r) * XW_COLS + outCol] = acc[r];
    }
}

// ---------------------------------------------------------------------------
// Self-loop contribution + bias; also initializes d_out (it is poisoned).
// out_mu[i][c]     = xw[i][c]    * dis[i]^2 + b_mu[c]
// out_logstd[i][c] = xw[i][64+c] * dis[i]^2 + b_logstd[c]
// ---------------------------------------------------------------------------
__global__ void k_selfloop_bias(const float* __restrict__ xw,
                                const float* __restrict__ dis,
                                const float* __restrict__ b_mu,
                                const float* __restrict__ b_ls,
                                float* __restrict__ out) {
    int idx = blockIdx.x * 256 + threadIdx.x;
    if (idx >= N_NODES * Z_DIM) return;
    int node = idx >> 6;
    int c    = idx & 63;
    float w = dis[node];
    w *= w;
    out[idx]                   = xw[(size_t)node * XW_COLS + c]        * w + b_mu[c];
    out[N_NODES * Z_DIM + idx] = xw[(size_t)node * XW_COLS + Z_DIM + c] * w + b_ls[c];
}

// ---------------------------------------------------------------------------
// Edge scatter: one wave per edge; lane handles cols {lane, lane+32} for both
// mu and logstd -> 4 global_atomic_add_f32 per lane. out fits in L2 (25.6MB
// of 192MB), so these atomics are L2-resident.
// ---------------------------------------------------------------------------
__global__ void k_edge_scatter(const float* __restrict__ xw,
                               const float* __restrict__ dis,
                               const long long* __restrict__ edge_index,
                               float* __restrict__ out) {
    const int lane = threadIdx.x & 31;
    const int wave = threadIdx.x >> 5;
    const int e    = blockIdx.x * 8 + wave;
    if (e >= N_EDGES) return;

    const int s = (int)edge_index[e];            // row 0 = src
    const int d = (int)edge_index[N_EDGES + e];  // row 1 = dst
    const float w = dis[s] * dis[d];

    const float* __restrict__ src = xw + (size_t)s * XW_COLS;
    float* __restrict__ mu = out + (size_t)d * Z_DIM;
    float* __restrict__ ls = out + (size_t)N_NODES * Z_DIM + (size_t)d * Z_DIM;

    const int c0 = lane;
    const int c1 = lane + 32;
    atomicAdd(&mu[c0], src[c0] * w);
    atomicAdd(&mu[c1], src[c1] * w);
    atomicAdd(&ls[c0], src[Z_DIM + c0] * w);
    atomicAdd(&ls[c1], src[Z_DIM + c1] * w);
}

// ---------------------------------------------------------------------------
// Launcher. Inputs: 0=x, 1=edge_index(int64), 2=W_mu, 3=b_mu, 4=W_logstd,
// 5=b_logstd. Output: [mu | logstd] fp32, 2*50000*64 elements.
// Workspace: xw (50000*128 f32 = 25.6MB) then dis (50000 f32).
// ---------------------------------------------------------------------------
extern "C" void kernel_launch(void* const* d_in, const int* in_sizes, int n_in,
                              void* d_out, int out_size, void* d_ws, size_t ws_size,
                              hipStream_t stream) {
    const float*     x    = (const float*)d_in[0];
    const long long* ei   = (const long long*)d_in[1];
    const float*     Wmu  = (const float*)d_in[2];
    const float*     bmu  = (const float*)d_in[3];
    const float*     Wls  = (const float*)d_in[4];
    const float*     bls  = (const float*)d_in[5];
    float*           out  = (float*)d_out;

    float* xw  = (float*)d_ws;                            // 50000*128 floats
    float* dis = xw + (size_t)N_NODES * XW_COLS;          // 50000 floats

    // degree / normalization
    k_init_deg<<<(N_NODES + 255) / 256, 256, 0, stream>>>(dis);
    k_count_deg<<<(N_EDGES + 255) / 256, 256, 0, stream>>>(ei, dis);
    k_rsqrt_deg<<<(N_NODES + 255) / 256, 256, 0, stream>>>(dis);

    // fp32 WMMA GEMM: 25000 tiles, 8 waves/block -> exact grid
    k_gemm_xw_wmma<<<TILES_M * TILES_N / 8, 256, 0, stream>>>(x, Wmu, Wls, xw);

    // init output with self-loop term + bias
    k_selfloop_bias<<<(N_NODES * Z_DIM + 255) / 256, 256, 0, stream>>>(
        xw, dis, bmu, bls, out);

    // scatter-add over edges: one wave per edge, exact grid
    k_edge_scatter<<<N_EDGES / 8, 256, 0, stream>>>(xw, dis, ei, out);
}